// MultiHeadAttention_69020124447162
// MI455X (gfx1250) — compile-verified
//
#include <hip/hip_runtime.h>
#include <math.h>

// ---------------------------------------------------------------------------
// Problem constants (B,S,D,H from the reference)
// ---------------------------------------------------------------------------
constexpr int Bb  = 4;
constexpr int Ss  = 2048;
constexpr int Dd  = 2048;
constexpr int Hh  = 16;
constexpr int DKk = 128;           // D / H
constexpr float SM_SCALE = 0.08838834764831845f;   // 1/sqrt(128)
constexpr float LN_EPS   = 1e-6f;

typedef float f32x2 __attribute__((ext_vector_type(2)));
typedef float f32x8 __attribute__((ext_vector_type(8)));
typedef int   i32x4 __attribute__((ext_vector_type(4)));

// ---------------------------------------------------------------------------
// Async global->LDS staging (CDNA5 GLOBAL_LOAD_ASYNC_TO_LDS_B128, ASYNCcnt).
// Builtin takes int4-typed pointers: (global i32x4*, lds i32x4*, imm, imm).
// Guarded so the file compiles on toolchains without the builtin.
// ---------------------------------------------------------------------------
#if defined(__has_builtin)
#if __has_builtin(__builtin_amdgcn_global_load_async_to_lds_b128)
#define HAVE_ASYNC_LDS 1
#endif
#endif

__device__ __forceinline__ void stage16(const float* __restrict__ g,
                                        float* __restrict__ l)
{
#ifdef HAVE_ASYNC_LDS
    __builtin_amdgcn_global_load_async_to_lds_b128(
        (__attribute__((address_space(1))) i32x4*)g,
        (__attribute__((address_space(3))) i32x4*)l,
        0, 0);
#else
    *reinterpret_cast<float4*>(l) = *reinterpret_cast<const float4*>(g);
#endif
}

__device__ __forceinline__ void stage_fence()
{
#ifdef HAVE_ASYNC_LDS
#if __has_builtin(__builtin_amdgcn_s_wait_asynccnt)
    __builtin_amdgcn_s_wait_asynccnt(0);
#else
    asm volatile("s_wait_asynccnt 0x0" ::: "memory");
#endif
#endif
}

// ---------------------------------------------------------------------------
// Tiled WMMA fp32 GEMM core:  C = alpha * A x op(B) (+ R)
//   A:  M x K, row-major, leading dim lda
//   B_TRANS=true : B is N x K row-major (NT gemm, "x @ W^T")
//   B_TRANS=false: B is K x N row-major (NN gemm, "attn @ V")
// Block tile 128x128, 256 threads (8 waves), wave tile 32x64 (2x4 WMMA tiles).
// Uses V_WMMA_F32_16X16X4_F32 (native fp32 matrix pipe, wave32).
//
// LDS strides chosen to be BOTH 16-byte aligned (for b128 async copies) and
// bank-conflict-free for the fragment read patterns:
//   LDS_A   = 36 floats: gcd(36,64)=4 -> 16 consecutive rows hit 16 distinct
//             banks; the +2 K-offset of lanes 16-31 cannot alias lanes 0-15
//             (36*dm === 2 mod 64 has no solution).
//   LDS_BNN = 136 floats: 2-row offset shifts banks by 16 -> disjoint from
//             the 16-bank window of lanes 0-15.
// ---------------------------------------------------------------------------
constexpr int BM = 128, BN = 128, BK = 32;
constexpr int LDS_A   = 36;
constexpr int LDS_BNN = 136;

template <bool B_TRANS, bool ADD_RES>
__device__ __forceinline__ void gemm_tile(
    const float* __restrict__ A,  int lda,
    const float* __restrict__ Bm, int ldb,
    float* __restrict__ C,        int ldc,
    const float* __restrict__ R,  int ldr,
    int K, float alpha, int bm, int bn)
{
    __shared__ __align__(16) float As[BM * LDS_A];
    __shared__ __align__(16) float Bs[B_TRANS ? (BN * LDS_A) : (BK * LDS_BNN)];

    const int tid  = threadIdx.x;     // 0..255
    const int lane = tid & 31;
    const int wave = tid >> 5;        // 0..7
    const int wm   = wave & 3;        // 4 waves along M (32 rows each)
    const int wn   = wave >> 2;       // 2 waves along N (64 cols each)
    const int lh   = lane & 15;       // row/col within 16
    const int kq   = (lane >> 4) << 1;     // K sub-offset: lanes 16-31 hold K+2,K+3
    const int rsel = (lane >> 4) << 3;     // C rows: lanes 16-31 hold M+8..M+15

    f32x8 acc[2][4];
#pragma unroll
    for (int mi = 0; mi < 2; ++mi)
#pragma unroll
        for (int ni = 0; ni < 4; ++ni)
            acc[mi][ni] = f32x8{0.f, 0.f, 0.f, 0.f, 0.f, 0.f, 0.f, 0.f};

    for (int k0 = 0; k0 < K; k0 += BK) {
        // ---- stage A tile: BM x BK (one b128 per thread per row-group) ----
        {
            int row = tid >> 3;
            const int col = (tid & 7) << 2;
#pragma unroll
            for (int r = 0; r < 4; ++r) {
                stage16(A + (size_t)(bm + row) * lda + k0 + col,
                        &As[row * LDS_A + col]);
                row += 32;
            }
        }
        // ---- stage B tile ----
        if (B_TRANS) {                       // B is N x K: tile BN x BK
            int row = tid >> 3;
            const int col = (tid & 7) << 2;
#pragma unroll
            for (int r = 0; r < 4; ++r) {
                stage16(Bm + (size_t)(bn + row) * ldb + k0 + col,
                        &Bs[row * LDS_A + col]);
                row += 32;
            }
        } else {                             // B is K x N: tile BK x BN
            int row = tid >> 5;              // k within tile
            const int col = (tid & 31) << 2; // n within tile
#pragma unroll
            for (int r = 0; r < 4; ++r) {
                stage16(Bm + (size_t)(k0 + row) * ldb + bn + col,
                        &Bs[row * LDS_BNN + col]);
                row += 8;
            }
        }
        stage_fence();        // drain ASYNCcnt before making tiles visible
        __syncthreads();

        // ---- compute: 8 k-steps of 4, 8 WMMAs per step ----
#pragma unroll
        for (int kk = 0; kk < BK; kk += 4) {
            f32x2 af[2], bf[4];
#pragma unroll
            for (int mi = 0; mi < 2; ++mi) {
                const int m = wm * 32 + mi * 16 + lh;
                af[mi].x = As[m * LDS_A + kk + kq];
                af[mi].y = As[m * LDS_A + kk + kq + 1];
            }
#pragma unroll
            for (int ni = 0; ni < 4; ++ni) {
                const int n = wn * 64 + ni * 16 + lh;
                if (B_TRANS) {
                    bf[ni].x = Bs[n * LDS_A + kk + kq];
                    bf[ni].y = Bs[n * LDS_A + kk + kq + 1];
                } else {
                    bf[ni].x = Bs[(kk + kq) * LDS_BNN + n];
                    bf[ni].y = Bs[(kk + kq + 1) * LDS_BNN + n];
                }
            }
#pragma unroll
            for (int mi = 0; mi < 2; ++mi)
#pragma unroll
                for (int ni = 0; ni < 4; ++ni)
                    acc[mi][ni] = __builtin_amdgcn_wmma_f32_16x16x4_f32(
                        false, af[mi], false, bf[ni],
                        (short)0, acc[mi][ni], false, false);
        }
        __syncthreads();
    }

    // ---- epilogue: scale, optional residual, store ----
#pragma unroll
    for (int mi = 0; mi < 2; ++mi) {
#pragma unroll
        for (int ni = 0; ni < 4; ++ni) {
            const int col = bn + wn * 64 + ni * 16 + lh;
#pragma unroll
            for (int i = 0; i < 8; ++i) {
                const int row = bm + wm * 32 + mi * 16 + i + rsel;
                float v = acc[mi][ni][i] * alpha;
                if (ADD_RES) v += R[(size_t)row * ldr + col];
                C[(size_t)row * ldc + col] = v;
            }
        }
    }
}

// ---------------------------------------------------------------------------
// Kernel wrappers
// ---------------------------------------------------------------------------

// Qp/Kp/Vp = X (8192x2048) @ W^T (2048x2048)
__global__ __launch_bounds__(256) void proj_kernel(
    const float* __restrict__ X, const float* __restrict__ W,
    float* __restrict__ Y)
{
    gemm_tile<true, false>(X, Dd, W, Dd, Y, Dd, nullptr, 0,
                           Dd, 1.0f, blockIdx.y * BM, blockIdx.x * BN);
}

// scores[n] = Qn @ Kn^T * scale, n = h*B + b, written into attn region
__global__ __launch_bounds__(256) void scores_kernel(
    const float* __restrict__ Qp, const float* __restrict__ Kp,
    float* __restrict__ attn)
{
    const int n = blockIdx.z;
    const int b = n & (Bb - 1);
    const int h = n >> 2;                       // n / B
    const float* Aq = Qp + ((size_t)b * Ss) * Dd + (size_t)h * DKk;
    const float* Bk = Kp + ((size_t)b * Ss) * Dd + (size_t)h * DKk;
    float* Cs = attn + (size_t)n * Ss * Ss;
    gemm_tile<true, false>(Aq, Dd, Bk, Dd, Cs, Ss, nullptr, 0,
                           DKk, SM_SCALE, blockIdx.y * BM, blockIdx.x * BN);
}

// row softmax in place: one 256-thread block per row of 2048
__global__ __launch_bounds__(256) void softmax_kernel(float* __restrict__ attn)
{
    float* p = attn + (size_t)blockIdx.x * Ss;
    const int tid = threadIdx.x;
    __shared__ float red[256];

    float v[8];
    float mx = -3.402823466e+38f;
#pragma unroll
    for (int i = 0; i < 8; ++i) { v[i] = p[tid + i * 256]; mx = fmaxf(mx, v[i]); }
    red[tid] = mx; __syncthreads();
    for (int s = 128; s > 0; s >>= 1) {
        if (tid < s) red[tid] = fmaxf(red[tid], red[tid + s]);
        __syncthreads();
    }
    mx = red[0]; __syncthreads();

    float sum = 0.f;
#pragma unroll
    for (int i = 0; i < 8; ++i) { v[i] = __expf(v[i] - mx); sum += v[i]; }
    red[tid] = sum; __syncthreads();
    for (int s = 128; s > 0; s >>= 1) {
        if (tid < s) red[tid] += red[tid + s];
        __syncthreads();
    }
    const float inv = 1.0f / red[0];
#pragma unroll
    for (int i = 0; i < 8; ++i) p[tid + i * 256] = v[i] * inv;
}

// context[n] = attn[n] @ Vn ; stored with the reference's batch-major
// reinterpretation: out row-block (b2 = n/H), column block (h2 = n%H)
__global__ __launch_bounds__(256) void ctx_kernel(
    const float* __restrict__ attn, const float* __restrict__ Vp,
    float* __restrict__ ctx)
{
    const int n  = blockIdx.z;
    const int b  = n & (Bb - 1);   // source head index n = h*B + b
    const int h  = n >> 2;
    const int b2 = n >> 4;         // reinterpretation: n = b2*H + h2
    const int h2 = n & (Hh - 1);
    const float* Aa = attn + (size_t)n * Ss * Ss;
    const float* Bv = Vp + ((size_t)b * Ss) * Dd + (size_t)h * DKk;
    float* Cc = ctx + ((size_t)b2 * Ss) * Dd + (size_t)h2 * DKk;
    gemm_tile<false, false>(Aa, Ss, Bv, Dd, Cc, Dd, nullptr, 0,
                            Ss, 1.0f, blockIdx.y * BM, blockIdx.x * BN);
}

// pre-LN output = Ctx @ Wo^T + residual(q)
__global__ __launch_bounds__(256) void outproj_kernel(
    const float* __restrict__ ctx, const float* __restrict__ Wo,
    const float* __restrict__ resid, float* __restrict__ Y)
{
    gemm_tile<true, true>(ctx, Dd, Wo, Dd, Y, Dd, resid, Dd,
                          Dd, 1.0f, blockIdx.y * BM, blockIdx.x * BN);
}

// LayerNorm over last dim (2048): one block per row
__global__ __launch_bounds__(256) void layernorm_kernel(
    const float* __restrict__ X, const float* __restrict__ gamma,
    const float* __restrict__ beta, float* __restrict__ out)
{
    const float* p = X + (size_t)blockIdx.x * Dd;
    float* o = out + (size_t)blockIdx.x * Dd;
    const int tid = threadIdx.x;
    __shared__ float red[256];

    float v[8];
    float s = 0.f;
#pragma unroll
    for (int i = 0; i < 8; ++i) { v[i] = p[tid + i * 256]; s += v[i]; }
    red[tid] = s; __syncthreads();
    for (int t = 128; t > 0; t >>= 1) {
        if (tid < t) red[tid] += red[tid + t];
        __syncthreads();
    }
    const float mean = red[0] * (1.0f / Dd);
    __syncthreads();

    float s2 = 0.f;
#pragma unroll
    for (int i = 0; i < 8; ++i) { const float d = v[i] - mean; s2 += d * d; }
    red[tid] = s2; __syncthreads();
    for (int t = 128; t > 0; t >>= 1) {
        if (tid < t) red[tid] += red[tid + t];
        __syncthreads();
    }
    const float inv = rsqrtf(red[0] * (1.0f / Dd) + LN_EPS);
#pragma unroll
    for (int i = 0; i < 8; ++i) {
        const int c = tid + i * 256;
        o[c] = (v[i] - mean) * inv * gamma[c] + beta[c];
    }
}

// ---------------------------------------------------------------------------
// Launcher
// d_in order: q, k, v, Wq, Wk, Wv, Wo, gamma, beta (all fp32)
// d_out: [output (B,S,D)] ++ [attn (H*B,S,S)]   (fp32)
// d_ws:  3 slabs of B*S*D floats (192 MB), liveness-reused
// ---------------------------------------------------------------------------
extern "C" void kernel_launch(void* const* d_in, const int* in_sizes, int n_in,
                              void* d_out, int out_size, void* d_ws, size_t ws_size,
                              hipStream_t stream)
{
    const float* q     = (const float*)d_in[0];
    const float* k     = (const float*)d_in[1];
    const float* v     = (const float*)d_in[2];
    const float* Wq    = (const float*)d_in[3];
    const float* Wk    = (const float*)d_in[4];
    const float* Wv    = (const float*)d_in[5];
    const float* Wo    = (const float*)d_in[6];
    const float* gamma = (const float*)d_in[7];
    const float* beta  = (const float*)d_in[8];

    float* out  = (float*)d_out;
    float* attn = out + (size_t)Bb * Ss * Dd;          // attn region of d_out

    const size_t slabElems = (size_t)Bb * Ss * Dd;     // 16,777,216 floats
    float* slab0 = (float*)d_ws;                       // Qp  -> later ctx
    float* slab1 = slab0 + slabElems;                  // Kp  -> later pre-LN out
    float* slab2 = slab1 + slabElems;                  // Vp

    const dim3 blk(256);
    const dim3 gProj(Dd / BN, (Bb * Ss) / BM);         // (16, 64)

    // 1) projections
    proj_kernel<<<gProj, blk, 0, stream>>>(q, Wq, slab0);
    proj_kernel<<<gProj, blk, 0, stream>>>(k, Wk, slab1);
    proj_kernel<<<gProj, blk, 0, stream>>>(v, Wv, slab2);

    // 2) raw scores straight into attn output region
    scores_kernel<<<dim3(Ss / BN, Ss / BM, Hh * Bb), blk, 0, stream>>>(slab0, slab1, attn);

    // 3) softmax in place (attn is a required output)
    softmax_kernel<<<dim3(Hh * Bb * Ss), blk, 0, stream>>>(attn);

    // 4) context (Qp slab is dead -> reuse as ctx)
    ctx_kernel<<<dim3(DKk / BN, Ss / BM, Hh * Bb), blk, 0, stream>>>(attn, slab2, slab0);

    // 5) output projection + residual (Kp slab dead -> pre-LN buffer)
    outproj_kernel<<<dim3(Dd / BN, (Bb * Ss) / BM), blk, 0, stream>>>(slab0, Wo, q, slab1);

    // 6) layernorm into final output
    layernorm_kernel<<<dim3(Bb * Ss), blk, 0, stream>>>(slab1, gamma, beta, out);
}